// ParallelScan_37374805409922
// MI455X (gfx1250) — compile-verified
//
#include <hip/hip_runtime.h>
#include <stdint.h>

// h[t] = a[t]*h[t-1] + b[t] over axis S for (B,S,D,N) = (2,2048,1024,16) f32.
// Memory-bound (768 MB @ 23.3 TB/s ~ 33us). Strategy: async global->LDS
// prefetch (ASYNCcnt) with a 3-deep ring buffer to keep ~16 MB in flight
// device-wide, decoupling HBM latency from the serial per-chain FMA chain.

#define B_DIM   2
#define S_LEN   2048
#define C_DIM   16384                      // D*N contiguous columns per timestep
#define COLS    128                        // columns per workgroup (== blockDim.x)
#define TSTEPS  32                         // timesteps per LDS tile
#define NTILES  (S_LEN / TSTEPS)           // 64
#define NBUF    3                          // ring depth: compute t, t+1/t+2 in flight
#define WAVE_SZ 32
#define NWAVES  (COLS / WAVE_SZ)           // 4 waves per block
#define ROWS_PER_WAVE (TSTEPS / NWAVES)    // 8 tile rows issued per wave
#define LOADS_PER_TILE (ROWS_PER_WAVE * 2) // 16 async b128 loads per wave per tile

typedef int v4i __attribute__((ext_vector_type(4)));
typedef __attribute__((address_space(1))) v4i* g_v4i_ptr;   // global
typedef __attribute__((address_space(3))) v4i* l_v4i_ptr;   // LDS

__device__ __forceinline__ void async_load_b128(const float* gsrc, float* ldst) {
#if __has_builtin(__builtin_amdgcn_global_load_async_to_lds_b128)
  __builtin_amdgcn_global_load_async_to_lds_b128(
      (g_v4i_ptr)(uintptr_t)gsrc,
      (l_v4i_ptr)(uint32_t)(uintptr_t)ldst,  // generic LDS addr low 32 bits == LDS offset
      0, /*cpol=*/0);
#else
  asm volatile("global_load_async_to_lds_b128 %0, %1, off"
               :
               : "v"((uint32_t)(uintptr_t)ldst), "v"((uint64_t)(uintptr_t)gsrc)
               : "memory");
#endif
}

template <int N>
__device__ __forceinline__ void wait_async() {
#if __has_builtin(__builtin_amdgcn_s_wait_asynccnt)
  __builtin_amdgcn_s_wait_asynccnt(N);
#else
  asm volatile("s_wait_asynccnt %0" ::"n"(N));
#endif
}

__global__ __launch_bounds__(COLS) void scan_async_kernel(
    const float* __restrict__ A, const float* __restrict__ Bv,
    float* __restrict__ H) {
  __shared__ float aT[NBUF][TSTEPS * COLS];  // 3 x 16KB
  __shared__ float bT[NBUF][TSTEPS * COLS];  // 3 x 16KB  (96KB total)

  const int tid  = threadIdx.x;              // == column within block's slab
  const int wave = tid >> 5;
  const int lane = tid & (WAVE_SZ - 1);

  const int blocksPerBatch = C_DIM / COLS;   // 128
  const int batch = blockIdx.x / blocksPerBatch;
  const int c0    = (blockIdx.x % blocksPerBatch) * COLS;

  const size_t base  = (size_t)batch * S_LEN * C_DIM + (size_t)c0;
  const float* aBase = A + base;
  const float* bBase = Bv + base;
  float*       hBase = H + base;

  // One b128 per lane covers a full 512B tile row; wave w owns rows
  // [w*8, w*8+8). 16 async loads per wave per tile, retired in order.
  auto issue_tile = [&](int tile, int buf) {
    const size_t gtile = (size_t)tile * TSTEPS * (size_t)C_DIM;
#pragma unroll
    for (int r = 0; r < ROWS_PER_WAVE; ++r) {
      const int row     = wave * ROWS_PER_WAVE + r;
      const size_t goff = gtile + (size_t)row * C_DIM + (size_t)(lane * 4);
      const int loff    = row * COLS + lane * 4;
      async_load_b128(aBase + goff, &aT[buf][loff]);
      async_load_b128(bBase + goff, &bT[buf][loff]);
    }
  };

  float h = 0.0f;                            // h[-1] = 0
  issue_tile(0, 0);                          // prologue: 2 tiles in flight
  issue_tile(1, 1);

  int buf = 0;
  for (int t = 0; t < NTILES; ++t) {
    if (t + 2 < NTILES) {
      // reuses tile (t-1)'s buffer; safe: trailing barrier of iter t-1 passed
      issue_tile(t + 2, (buf + 2 >= NBUF) ? buf + 2 - NBUF : buf + 2);
    }
    // In-order ASYNCcnt retirement: <= 2 tiles outstanding => tile t landed.
    const int rem = NTILES - 1 - t;
    if (rem >= 2)      wait_async<2 * LOADS_PER_TILE>();
    else if (rem == 1) wait_async<LOADS_PER_TILE>();
    else               wait_async<0>();
    __syncthreads();                         // all waves' tile-t rows visible

    const size_t srow = (size_t)t * TSTEPS;
#pragma unroll
    for (int s = 0; s < TSTEPS; ++s) {
      const float av = aT[buf][s * COLS + tid];  // conflict-free LDS
      const float bv = bT[buf][s * COLS + tid];
      h = __builtin_fmaf(av, h, bv);
      __builtin_nontemporal_store(h, hBase + (srow + s) * C_DIM + tid);
    }
    __syncthreads();                         // tile-t buffer free for reuse

    buf = (buf + 1 >= NBUF) ? 0 : buf + 1;
  }
}

extern "C" void kernel_launch(void* const* d_in, const int* in_sizes, int n_in,
                              void* d_out, int out_size, void* d_ws, size_t ws_size,
                              hipStream_t stream) {
  (void)in_sizes; (void)n_in; (void)out_size; (void)d_ws; (void)ws_size;
  const float* a = (const float*)d_in[0];
  const float* b = (const float*)d_in[1];
  float*       h = (float*)d_out;

  dim3 grid(B_DIM * (C_DIM / COLS));  // 256 blocks
  dim3 block(COLS);                   // 128 threads = 4 waves
  hipLaunchKernelGGL(scan_async_kernel, grid, block, 0, stream, a, b, h);
}